// GumbelSinkhorn_3848290697878
// MI455X (gfx1250) — compile-verified
//
#include <hip/hip_runtime.h>
#include <cmath>
#include <cstdint>

typedef __attribute__((ext_vector_type(2))) float v2f;
typedef __attribute__((ext_vector_type(8))) float v8f;

#define NMAT 128      // 8*16 independent matrices
#define N    512      // matrix dimension
#define TPAD 516      // padded LDS tile stride (floats) -> conflict-free banks

// ---------------- init: zero the column potentials ----------------
__global__ void gs_zero(float* __restrict__ p, int n) {
  int i = blockIdx.x * blockDim.x + threadIdx.x;
  if (i < n) p[i] = 0.0f;
}

// ---------------- row pass: r[b][i] = LSE_j( S[b][i][j] - c[b][j] ) ----------------
// grid = NMAT*32 blocks of 256 threads; each block owns a 16-row slab of one matrix.
// Tile staged into LDS via async global->LDS b128 copies; exp-sums reduced with
// chained v_wmma_f32_16x16x4_f32 against an all-ones B matrix.
__global__ __launch_bounds__(256) void gs_row(const float* __restrict__ S,
                                              const float* __restrict__ c,
                                              float* __restrict__ r) {
  __shared__ __align__(16) float tile[16 * TPAD];
  __shared__ __align__(16) float cs[N];
  __shared__ float wmax[8][16];
  __shared__ float wsum[8][16];
  __shared__ float rowmax[16];

  const int t   = threadIdx.x;
  const int blk = blockIdx.x;
  const int b   = blk >> 5;          // matrix index
  const int r0  = (blk & 31) << 4;   // first row of this 16-row slab

  const float* gbase = S + (size_t)b * (N * N) + (size_t)r0 * N;

  // stage c[b][0..511] (2 KB) with regular b128 loads
  if (t < N / 4) {
    ((float4*)cs)[t] = ((const float4*)(c + ((size_t)b << 9)))[t];
  }

  // async-stage the 16x512 tile: 2048 float4s, 8 per thread, fully coalesced
  for (int k = 0; k < 8; ++k) {
    unsigned q      = (unsigned)t + ((unsigned)k << 8);  // float4 index
    unsigned row    = q >> 7;                            // 128 float4 per row
    unsigned col4   = q & 127;
    unsigned ldsOff = (unsigned)(uintptr_t)&tile[row * TPAD + col4 * 4];
    const float* gp = gbase + row * N + col4 * 4;
    asm volatile("global_load_async_to_lds_b128 %0, %1, off"
                 :: "v"(ldsOff), "v"(gp) : "memory");
  }
  asm volatile("s_wait_asynccnt 0" ::: "memory");
  __syncthreads();

  const int wave = t >> 5;        // 0..7, owns 64-column slice
  const int lane = t & 31;
  const int m    = lane & 15;     // row within slab (WMMA A-matrix M index)
  const int khi  = lane >> 4;     // K-half (A layout: lanes>=16 hold K=2,3)
  const int jw   = wave << 6;

  // ---- phase 1: per-row max over this wave's 64 columns ----
  float mx = -INFINITY;
#pragma unroll
  for (int j0 = 0; j0 < 64; j0 += 4) {
    int j    = jw + j0 + (khi << 1);
    float x0 = tile[m * TPAD + j]     - cs[j];
    float x1 = tile[m * TPAD + j + 1] - cs[j + 1];
    mx = fmaxf(mx, fmaxf(x0, x1));
  }
  mx = fmaxf(mx, __shfl_xor(mx, 16, 32));   // combine the two K-halves (same row)
  if (lane < 16) wmax[wave][m] = mx;
  __syncthreads();
  if (t < 16) {
    float v = wmax[0][t];
#pragma unroll
    for (int w = 1; w < 8; ++w) v = fmaxf(v, wmax[w][t]);
    rowmax[t] = v;
  }
  __syncthreads();
  const float rm = rowmax[m];

  // ---- phase 2: sum of exp via chained WMMA (B = ones => D_i* = rowsum(A_i)) ----
  v8f acc = {0.f, 0.f, 0.f, 0.f, 0.f, 0.f, 0.f, 0.f};
  const v2f ones = {1.0f, 1.0f};
#pragma unroll
  for (int j0 = 0; j0 < 64; j0 += 4) {
    int j = jw + j0 + (khi << 1);
    v2f a;
    a.x = __expf(tile[m * TPAD + j]     - cs[j]     - rm);
    a.y = __expf(tile[m * TPAD + j + 1] - cs[j + 1] - rm);
    acc = __builtin_amdgcn_wmma_f32_16x16x4_f32(false, a, false, ones,
                                                (short)0, acc, false, false);
  }

  // D layout: VGPR v -> (M=v, N=lane) for lanes 0-15; (M=v+8, N=lane-16) for 16-31.
  // Row sums replicated across N, so lane 0 / lane 16 column 0 carry all 16 rows.
  if (lane == 0) {
#pragma unroll
    for (int vv = 0; vv < 8; ++vv) wsum[wave][vv] = acc[vv];
  } else if (lane == 16) {
#pragma unroll
    for (int vv = 0; vv < 8; ++vv) wsum[wave][8 + vv] = acc[vv];
  }
  __syncthreads();
  if (t < 16) {
    float s = 0.0f;
#pragma unroll
    for (int w = 0; w < 8; ++w) s += wsum[w][t];
    r[((size_t)b << 9) + r0 + t] = rowmax[t] + __logf(s);
  }
}

// ---------------- col pass: c[b][j] = LSE_i( S[b][i][j] - r[b][i] ) ----------------
// grid = NMAT blocks of 1024 threads; thread t owns column j = t&511 over half the
// rows (half = t>>9). Consecutive lanes touch consecutive j -> fully coalesced.
__global__ __launch_bounds__(1024) void gs_col(const float* __restrict__ S,
                                               const float* __restrict__ r,
                                               float* __restrict__ c) {
  __shared__ __align__(16) float rs[N];
  __shared__ float pm[N];
  __shared__ float ps[N];

  const int t = threadIdx.x;
  const int b = blockIdx.x;

  if (t < N / 4) {
    ((float4*)rs)[t] = ((const float4*)(r + ((size_t)b << 9)))[t];
  }
  __syncthreads();

  const int j    = t & (N - 1);
  const int half = t >> 9;
  const float* col = S + ((size_t)b << 18) + ((size_t)(half * 256) << 9) + j;

  float m = -INFINITY, s = 0.0f;
  for (int i = 0; i < 256; ++i) {
    float x  = col[(size_t)i << 9] - rs[half * 256 + i];
    float m2 = fmaxf(m, x);
    s = s * __expf(m - m2) + __expf(x - m2);
    m = m2;
  }

  if (half) { pm[j] = m; ps[j] = s; }
  __syncthreads();
  if (!half) {
    float m1 = pm[j], s1 = ps[j];
    float M  = fmaxf(m, m1);
    float Sv = s * __expf(m - M) + s1 * __expf(m1 - M);
    c[((size_t)b << 9) + j] = M + __logf(Sv);
  }
}

// ---------------- final: out = exp(S - r_i - c_j), float4 per thread ----------------
__global__ __launch_bounds__(256) void gs_final(const float* __restrict__ S,
                                                const float* __restrict__ r,
                                                const float* __restrict__ c,
                                                float* __restrict__ out) {
  const size_t idx4 = (size_t)blockIdx.x * 256 + threadIdx.x;  // float4 index
  const int b  = (int)(idx4 >> 16);        // 65536 float4 per matrix
  const int q  = (int)(idx4 & 65535);
  const int i  = q >> 7;
  const int j4 = q & 127;

  float4 s  = ((const float4*)S)[idx4];
  float  ri = r[(b << 9) + i];
  float4 c4 = ((const float4*)c)[(b << 7) + j4];

  float4 o;
  o.x = __expf(s.x - ri - c4.x);
  o.y = __expf(s.y - ri - c4.y);
  o.z = __expf(s.z - ri - c4.z);
  o.w = __expf(s.w - ri - c4.w);
  ((float4*)out)[idx4] = o;
}

extern "C" void kernel_launch(void* const* d_in, const int* in_sizes, int n_in,
                              void* d_out, int out_size, void* d_ws, size_t ws_size,
                              hipStream_t stream) {
  const float* S = (const float*)d_in[0];
  float* out = (float*)d_out;
  float* r = (float*)d_ws;          // NMAT*N floats
  float* c = r + NMAT * N;          // NMAT*N floats (total ws use: 512 KB)

  gs_zero<<<(NMAT * N + 255) / 256, 256, 0, stream>>>(c, NMAT * N);
  for (int it = 0; it < 10; ++it) {
    gs_row<<<NMAT * 32, 256, 0, stream>>>(S, c, r);
    gs_col<<<NMAT, 1024, 0, stream>>>(S, r, c);
  }
  gs_final<<<(NMAT * N * N / 4) / 256, 256, 0, stream>>>(S, r, c, out);
}